// Haar_38027640439240
// MI455X (gfx1250) — compile-verified
//
#include <hip/hip_runtime.h>
#include <math.h>

#define INV_SQRT2F 0.7071067811865476f

#define B_  16
#define T_  8192
#define C_  256
#define NX  33554432   // 16*8192*256
#define NH  16777216   // NX/2

#define TILE_C 64
#define TILE_T 128
#define LDS_STRIDE 132   // floats: 128 + 4 pad (multiple of 4 -> 16B aligned rows)

// ---------------- CDNA5 async global->LDS copy (b32 scatter) ----------------
// Builtin params are address-space-qualified int* (AS1 src, AS3 dst).
#if defined(__HIP_DEVICE_COMPILE__) && __has_builtin(__builtin_amdgcn_global_load_async_to_lds_b32)
#define ASYNC_B32(g, l)                                                        \
  __builtin_amdgcn_global_load_async_to_lds_b32(                               \
      (__attribute__((address_space(1))) int*)(int*)(void*)(g),                \
      (__attribute__((address_space(3))) int*)(int*)(void*)(l), 0, 0)
#else
__device__ __forceinline__ void async_b32_asm(const void* g, void* l) {
  unsigned lds_off =
      (unsigned)(unsigned long long)(__attribute__((address_space(3))) void*)l;
  asm volatile("global_load_async_to_lds_b32 %0, %1, off"
               :: "v"(lds_off), "v"(g) : "memory");
}
#define ASYNC_B32(g, l) async_b32_asm((const void*)(g), (void*)(l))
#endif

__device__ __forceinline__ void wait_async0() {
#if defined(__HIP_DEVICE_COMPILE__) && __has_builtin(__builtin_amdgcn_s_wait_asynccnt)
  __builtin_amdgcn_s_wait_asynccnt(0);
#else
  asm volatile("s_wait_asynccnt 0" ::: "memory");
#endif
}

// ---------------- main kernel: transpose + Haar + partial reductions --------
__global__ __launch_bounds__(256) void haar_kernel(const float* __restrict__ x,
                                                   float* __restrict__ out,
                                                   float* __restrict__ ws) {
  __shared__ __align__(16) float tile[TILE_C * LDS_STRIDE];

  const int tid = threadIdx.x;
  const int bid = blockIdx.x;
  const int ct = bid & 3;          // channel tile  (4)
  const int tt = (bid >> 2) & 63;  // time tile     (64)
  const int b  = bid >> 8;         // batch         (16)
  const int c0 = ct * TILE_C;
  const int t0 = tt * TILE_T;
  const int k0 = tt * (TILE_T / 2);

  // ---- load phase: coalesced global reads, async DMA scatter into tile[c][t]
  {
    const int cl = tid & 63;   // channel within tile (lanes = consecutive c)
    const int tb = tid >> 6;   // 0..3
    const float* gp = x + ((size_t)b * T_ + (size_t)(t0 + tb)) * C_ + (c0 + cl);
    float* lp = &tile[cl * LDS_STRIDE + tb];
#pragma unroll
    for (int i = 0; i < TILE_T / 4; ++i) {   // 32 async b32 ops per thread
      ASYNC_B32(gp, lp);
      gp += 4 * C_;
      lp += 4;
    }
  }
  wait_async0();
  __syncthreads();

  // ---- compute phase: b128 LDS reads, coalesced float2 global writes -------
  const int w = tid >> 5;  // wave 0..7
  const int l = tid & 31;  // lane 0..31 (wave32)
  float sA = 0.f, sX = 0.f, sD = 0.f;

  float* approx  = out;
  float* details = out + (NH + 1);

#pragma unroll
  for (int r8 = 0; r8 < 8; ++r8) {
    const int cc = w * 8 + r8;  // uniform per wave
    const float4 v = *(const float4*)&tile[cc * LDS_STRIDE + 4 * l];
    const float a0 = (v.x + v.y) * INV_SQRT2F;
    const float d0 = (v.x - v.y) * INV_SQRT2F;
    const float a1 = (v.z + v.w) * INV_SQRT2F;
    const float d1 = (v.z - v.w) * INV_SQRT2F;
    const size_t row = (size_t)(b * C_ + c0 + cc) * (T_ / 2) + (size_t)(k0 + 2 * l);
    *(float2*)&approx[row]  = make_float2(a0, a1);
    *(float2*)&details[row] = make_float2(d0, d1);
    sA += a0 + a1;
    sX += v.x + v.y + v.z + v.w;
    sD += fabsf(d0) + fabsf(d1);
  }

  // wave32 reduction, then 3 atomics per wave
#pragma unroll
  for (int off = 16; off > 0; off >>= 1) {
    sA += __shfl_down(sA, off, 32);
    sX += __shfl_down(sX, off, 32);
    sD += __shfl_down(sD, off, 32);
  }
  if (l == 0) {
    atomicAdd(&ws[0], sA);
    atomicAdd(&ws[1], sX);
    atomicAdd(&ws[2], sD);
  }
}

// ---------------- finalize: r = 0.01*|mean(approx)-mean(x)| + 0.01*mean|det|
__global__ void haar_finalize(const float* __restrict__ ws,
                              float* __restrict__ out) {
  const float meanA = ws[0] * (1.0f / (float)NH);
  const float meanX = ws[1] * (1.0f / (float)NX);
  const float meanD = ws[2] * (1.0f / (float)NH);
  out[NH] = 0.01f * fabsf(meanA - meanX) + 0.01f * meanD;
}

extern "C" void kernel_launch(void* const* d_in, const int* in_sizes, int n_in,
                              void* d_out, int out_size, void* d_ws, size_t ws_size,
                              hipStream_t stream) {
  (void)in_sizes; (void)n_in; (void)out_size; (void)ws_size;
  const float* x = (const float*)d_in[0];
  float* out = (float*)d_out;
  float* ws  = (float*)d_ws;

  (void)hipMemsetAsync(d_ws, 0, 3 * sizeof(float), stream);  // capture-safe memset node
  haar_kernel<<<4096, 256, 0, stream>>>(x, out, ws);
  haar_finalize<<<1, 1, 0, stream>>>(ws, out);
}